// ScaledDotProductAttention_3770981285895
// MI455X (gfx1250) — compile-verified
//
#include <hip/hip_runtime.h>

typedef float v2f __attribute__((ext_vector_type(2)));
typedef float v8f __attribute__((ext_vector_type(8)));

#define D_H   128
#define L_SEQ 2048
#define QT    16          // q rows per workgroup
#define SROW  2052        // padded score row stride (floats): 4*r mod 64 -> bank-conflict-free b64 reads
#define QROW  132         // padded q row stride (floats)
#define NWAVE 8
#define DROP_P 0.1f
#define KEEP_SCALE (1.0f/0.9f)

__global__ __launch_bounds__(256)
void attn_fp32_wmma_kernel(const float* __restrict__ q,
                           const float* __restrict__ k,
                           const float* __restrict__ v,
                           const float* __restrict__ u,
                           float* __restrict__ out,
                           float* __restrict__ attn)
{
    __shared__ float s_scores[QT * SROW];   // 16 x 2052 f32 = 128.25 KB
    __shared__ float s_q[QT * QROW];        // 16 x 132  f32 = 8.25 KB
    __shared__ float s_rinv[QT];

    const int t    = threadIdx.x;
    const int lane = t & 31;
    const int wave = t >> 5;
    const int hi16 = lane >> 4;     // 0 for lanes 0-15, 1 for lanes 16-31
    const int ln   = lane & 15;

    const int blk = blockIdx.x;     // 0 .. B*L/QT-1 = 2047
    const int bb  = blk >> 7;       // batch (L/QT = 128)
    const int qt  = blk & 127;
    const int q0  = qt * QT;

    const float* qb = q + (size_t)bb * L_SEQ * D_H;
    const float* kb = k + (size_t)bb * L_SEQ * D_H;
    const float* vb = v + (size_t)bb * L_SEQ * D_H;
    const float* ub = u + (size_t)bb * L_SEQ * L_SEQ;

    // ---- stage Q tile into LDS (coalesced) ----
    for (int e = t; e < QT * D_H; e += 256) {
        int r = e >> 7, c = e & 127;
        s_q[r * QROW + c] = qb[(size_t)(q0 + r) * D_H + c];
    }
    __syncthreads();

    const float inv_temper = 0.08838834764831845f; // 1/sqrt(128)

    // A-fragment source in LDS (per-lane, constant across k-tiles):
    // A 16x4 f32 layout: lane ln holds row M=ln; VGPR0 = K=2*hi16, VGPR1 = K=2*hi16+1
    const float* aq_base = &s_q[ln * QROW + 2 * hi16];

    // ---- phase 1: S = (Q K^T)/temper with pre-softmax dropout, kept in LDS ----
    for (int kt = wave; kt < L_SEQ / 16; kt += NWAVE) {
        const int n0 = kt * 16;
        v8f acc = {};
        // B 4x16 f32 layout: lane holds col N=ln; VGPR0 = K=2*hi16, VGPR1 = K=2*hi16+1
        const float* kcol = kb + (size_t)(n0 + ln) * D_H + 2 * hi16;
        #pragma unroll 8
        for (int s = 0; s < 32; ++s) {
            v2f afrag = *(const v2f*)(aq_base + 4 * s);   // ds_load_b64, conflict-free
            v2f bfrag = *(const v2f*)(kcol + 4 * s);      // global_load_b64, L2-resident K
            acc = __builtin_amdgcn_wmma_f32_16x16x4_f32(false, afrag, false, bfrag,
                                                        (short)0, acc, false, false);
        }
        // D layout: vgpr i -> (M = i + 8*hi16, N = ln)
        #pragma unroll
        for (int i = 0; i < 8; ++i) {
            int M = i + 8 * hi16;
            float uval = ub[(size_t)(q0 + M) * L_SEQ + n0 + ln];
            float sv = acc[i] * inv_temper * (uval >= DROP_P ? KEEP_SCALE : 0.0f);
            s_scores[M * SROW + n0 + ln] = sv;
        }
    }
    __syncthreads();

    // ---- phase 2: row softmax (16 threads per row, shuffle reductions) ----
    {
        const int r   = t >> 4;    // 0..15 (rows 2w, 2w+1 live in wave w's two 16-lane halves)
        const int sub = t & 15;
        float* row = &s_scores[r * SROW];
        float m = -3.402823466e38f;
        for (int c = sub; c < L_SEQ; c += 16) m = fmaxf(m, row[c]);
        #pragma unroll
        for (int o = 8; o > 0; o >>= 1) m = fmaxf(m, __shfl_xor(m, o, 16));
        float ssum = 0.0f;
        for (int c = sub; c < L_SEQ; c += 16) {
            float e = __expf(row[c] - m);
            row[c] = e;
            ssum += e;
        }
        #pragma unroll
        for (int o = 8; o > 0; o >>= 1) ssum += __shfl_xor(ssum, o, 16);
        if (sub == 0) s_rinv[r] = 1.0f / ssum;
    }
    __syncthreads();

    // ---- phase 3: normalize in LDS + stream attn to global (coalesced, written once) ----
    float* attn_b = attn + (size_t)bb * L_SEQ * L_SEQ;
    for (int e = t; e < QT * L_SEQ; e += 256) {
        int r = e >> 11;
        int c = e & (L_SEQ - 1);
        float pval = s_scores[r * SROW + c] * s_rinv[r];
        s_scores[r * SROW + c] = pval;
        attn_b[(size_t)(q0 + r) * L_SEQ + c] = pval;
    }
    __syncthreads();

    // ---- phase 4: O = P @ V, each wave owns one 16-wide output column tile ----
    {
        const int n0 = wave * 16;
        v8f acc = {};
        const float* vcol = vb + n0 + ln;
        const float* prow_base = &s_scores[ln * SROW + 2 * hi16];
        for (int kk = 0; kk < L_SEQ; kk += 4) {
            // A fragment from LDS scores: conflict-free b64 (row stride 2052)
            v2f afrag = *(const v2f*)(prow_base + kk);
            v2f bfrag;
            bfrag.x = vcol[(size_t)(kk +     2 * hi16) * D_H];
            bfrag.y = vcol[(size_t)(kk + 1 + 2 * hi16) * D_H];
            acc = __builtin_amdgcn_wmma_f32_16x16x4_f32(false, afrag, false, bfrag,
                                                        (short)0, acc, false, false);
        }
        float* ob = out + (size_t)bb * L_SEQ * D_H + (size_t)q0 * D_H;
        #pragma unroll
        for (int i = 0; i < 8; ++i) {
            int M = i + 8 * hi16;
            ob[(size_t)M * D_H + n0 + ln] = acc[i];
        }
    }
}

extern "C" void kernel_launch(void* const* d_in, const int* in_sizes, int n_in,
                              void* d_out, int out_size, void* d_ws, size_t ws_size,
                              hipStream_t stream) {
    const float* q = (const float*)d_in[0];
    const float* k = (const float*)d_in[1];
    const float* v = (const float*)d_in[2];
    const float* u = (const float*)d_in[3];

    float* out  = (float*)d_out;                       // [16, 2048, 128]
    float* attn = out + (size_t)16 * L_SEQ * D_H;      // [16, 2048, 2048]

    dim3 grid(16 * (L_SEQ / QT));   // 2048 workgroups: one per (batch, 16-row q block)
    dim3 block(256);                // 8 wave32 waves
    attn_fp32_wmma_kernel<<<grid, block, 0, stream>>>(q, k, v, u, out, attn);
}